// JointMembership_QuantumCircuit_89816356094178
// MI455X (gfx1250) — compile-verified
//
#include <hip/hip_runtime.h>
#include <math.h>

// ---------------------------------------------------------------------------
// Memory-bound streaming kernel: 128 MiB total traffic -> ~5.6 us floor at
// 23.3 TB/s. Closed-form outputs:
//   out0 = 0.5 + 0.5*(cos(theta)*cos(x0) - sin(theta)*sin(x0)*sin(x1))
//   out1 = 0.5 + 0.5*(cos(x0)*cos(x1))
// (exact algebraic reduction of the reference circuit). Per-element cost is
// 2 fast sincos (~20 VALU each) + ~6 FMAs, keeping dynamic VALU at/below the
// HBM roofline. Data movement uses the CDNA5 async global->LDS path
// (ASYNCcnt) with double buffering, plus non-temporal b128 stores.
// ---------------------------------------------------------------------------

typedef float v4f __attribute__((ext_vector_type(4)));
// Async builtin params are pointers to gcc-style int vectors
// (per hipcc diagnostic: 'int __vector(4) __device__ *').
typedef int qvi4 __attribute__((vector_size(16)));

#if defined(__HIP_DEVICE_COMPILE__) && \
    __has_builtin(__builtin_amdgcn_global_load_async_to_lds_b128) && \
    __has_builtin(__builtin_amdgcn_s_wait_asynccnt)
#define QC_HAS_ASYNC 1
#else
#define QC_HAS_ASYNC 0
#endif

#if QC_HAS_ASYNC
__device__ __forceinline__ void qc_async_copy_b128(const float* gsrc, float* ldst) {
  // Per-lane: copy 16 bytes global -> LDS via the async DMA path (ASYNCcnt).
  __builtin_amdgcn_global_load_async_to_lds_b128(
      (__attribute__((address_space(1))) qvi4*)gsrc,
      (__attribute__((address_space(3))) qvi4*)ldst,
      /*imm offset=*/0, /*cpol=*/0);
}
#endif

// Fast sincos: 3-term Cody-Waite pi/2 reduction (FMA) + cephes minimax polys.
// Accurate to ~1e-7 for |x| up to a few thousand; inputs here are N(0,pi)
// samples (|x| < ~20). ~20 VALU ops, fully branchless quadrant selection.
__device__ __forceinline__ void fast_sincosf(float x, float& s, float& c) {
  const float TWO_OVER_PI = 0.63661977236758134f;
  const float DP1 = 1.5703125f;                 // pi/2 split, high
  const float DP2 = 4.837512969970703125e-4f;   // pi/2 split, mid
  const float DP3 = 7.549789948768648e-8f;      // pi/2 split, low
  float fn = rintf(x * TWO_OVER_PI);
  int k = (int)fn;
  float r = fmaf(fn, -DP1, x);
  r = fmaf(fn, -DP2, r);
  r = fmaf(fn, -DP3, r);
  float r2 = r * r;
  // sin(r) = r + r^3 * ps(r^2), cos(r) = 1 - r^2/2 + r^4 * pc(r^2)
  float ps = fmaf(r2, fmaf(r2, -1.9515295891e-4f, 8.3321608736e-3f),
                  -1.6666654611e-1f);
  float sr = fmaf(r * r2, ps, r);
  float pc = fmaf(r2, fmaf(r2, 2.443315711809948e-5f, -1.388731625493765e-3f),
                  4.166664568298827e-2f);
  float cr = fmaf(r2 * r2, pc, fmaf(r2, -0.5f, 1.0f));
  // quadrant: x = r + k*pi/2
  bool swap = (k & 1) != 0;
  float ss = swap ? cr : sr;
  float cc = swap ? sr : cr;
  if (k & 2) ss = -ss;
  if ((k + 1) & 2) cc = -cc;
  s = ss;
  c = cc;
}

// One batch element via the closed form (needs full-angle sincos).
__device__ __forceinline__ void qc_elem(float x0, float x1, float cosT, float sinT,
                                        float& o0, float& o1) {
  float sx0, cx0, sx1, cx1;
  fast_sincosf(x0, sx0, cx0);
  fast_sincosf(x1, sx1, cx1);
  float t = sinT * sx0;
  o0 = fmaf(0.5f, fmaf(cosT, cx0, -(t * sx1)), 0.5f);
  o1 = fmaf(0.5f * cx0, cx1, 0.5f);
}

// Pipelined kernel: 256 threads/block (8 wave32s). Tile = 512 batch elements
// (4 KiB in / 4 KiB out per block-tile). Double-buffered async LDS staging:
// prefetch tile t+1 while computing tile t. Each lane reads back only the LDS
// bytes written by its own wave's async instruction, so per-wave
// s_wait_asynccnt is the only synchronization required (no barriers).
__global__ __launch_bounds__(256) void qc_pipe(const float* __restrict__ x,
                                               const float* __restrict__ theta,
                                               float* __restrict__ out,
                                               int tilesPerBlock) {
#if QC_HAS_ASYNC
  __shared__ __align__(16) float lds[2][1024];  // 2 x 4 KiB double buffer
#endif
  const int tid = threadIdx.x;

  // Uniform trainable angle: precise OCML sincos of theta/2, combined via
  // double-angle identities (matches reference's half-angle path). This is
  // uniform -> compiler scalarizes it onto the SALU, runs once per wave.
  float sh, ch;
  sincosf(theta[0] * 0.5f, &sh, &ch);
  const float cosT = ch * ch - sh * sh;  // cos(theta)
  const float sinT = 2.0f * ch * sh;     // sin(theta)

  const long long blockBase = (long long)blockIdx.x * tilesPerBlock * 512;

#if QC_HAS_ASYNC
  // Prologue: issue async copy of tile 0 into buffer 0.
  qc_async_copy_b128(x + blockBase * 2 + tid * 4, &lds[0][tid * 4]);
#endif

  for (int t = 0; t < tilesPerBlock; ++t) {
    const long long ebase = blockBase + (long long)t * 512;  // batch elem base
    v4f xi;
#if QC_HAS_ASYNC
    const int cur = t & 1;
    if (t + 1 < tilesPerBlock) {
      // Prefetch next tile into the other buffer, then retire the oldest.
      qc_async_copy_b128(x + (ebase + 512) * 2 + tid * 4, &lds[cur ^ 1][tid * 4]);
      __builtin_amdgcn_s_wait_asynccnt(1);
    } else {
      __builtin_amdgcn_s_wait_asynccnt(0);
    }
    asm volatile("" ::: "memory");  // compiler barrier: LDS reads stay below wait
    xi = *(const v4f*)&lds[cur][tid * 4];
#else
    xi = *(const v4f*)(x + ebase * 2 + tid * 4);
#endif
    float o0, o1, o2, o3;
    qc_elem(xi.x, xi.y, cosT, sinT, o0, o1);  // batch element ebase + tid*2
    qc_elem(xi.z, xi.w, cosT, sinT, o2, o3);  // batch element ebase + tid*2 + 1
    v4f o = {o0, o1, o2, o3};
    // Streaming output, never re-read: non-temporal b128 store.
    __builtin_nontemporal_store(o, (v4f*)(out + ebase * 2 + tid * 4));
  }
}

// Scalar fallback for sizes not divisible by the block tile.
__global__ __launch_bounds__(256) void qc_simple(const float* __restrict__ x,
                                                 const float* __restrict__ theta,
                                                 float* __restrict__ out,
                                                 long long n) {
  long long i = (long long)blockIdx.x * blockDim.x + threadIdx.x;
  if (i >= n) return;
  float sh, ch;
  sincosf(theta[0] * 0.5f, &sh, &ch);
  float o0, o1;
  qc_elem(x[2 * i], x[2 * i + 1], ch * ch - sh * sh, 2.0f * ch * sh, o0, o1);
  out[2 * i] = o0;
  out[2 * i + 1] = o1;
}

extern "C" void kernel_launch(void* const* d_in, const int* in_sizes, int n_in,
                              void* d_out, int out_size, void* d_ws, size_t ws_size,
                              hipStream_t stream) {
  const float* x = (const float*)d_in[0];      // [B,2] float32
  const float* theta = (const float*)d_in[1];  // [1]   float32
  float* out = (float*)d_out;                  // [B,2] float32

  const long long n = (long long)in_sizes[0] / 2;  // batch count B
  const int TILES_PER_BLOCK = 8;
  const long long ELEMS_PER_BLOCK = 512LL * TILES_PER_BLOCK;  // 4096

  if (n > 0 && (n % ELEMS_PER_BLOCK) == 0) {
    const unsigned grid = (unsigned)(n / ELEMS_PER_BLOCK);  // 2048 for B=2^23
    qc_pipe<<<grid, 256, 0, stream>>>(x, theta, out, TILES_PER_BLOCK);
  } else if (n > 0) {
    const unsigned grid = (unsigned)((n + 255) / 256);
    qc_simple<<<grid, 256, 0, stream>>>(x, theta, out, n);
  }
}